// Conv_with_nebs_81922206204025
// MI455X (gfx1250) — compile-verified
//
#include <hip/hip_runtime.h>
#include <hip/hip_bf16.h>
#include <cstdint>
#include <cstddef>

// ---------------------------------------------------------------------------
// Spherical CNN forward on MI455X (gfx1250, wave32).
// f32 WMMA (V_WMMA_F32_16X16X4_F32) for conv2 / conv3 / fc GEMMs.
// Workload is HBM-bound on the 335MB Wfc stream (~15us at 23.3TB/s), so f32
// precision is free; WMMA fragments are v2f (16x4 / 4x16 tiles per wave).
// ---------------------------------------------------------------------------

typedef float v2f __attribute__((ext_vector_type(2)));
typedef float v8f __attribute__((ext_vector_type(8)));

#define NVERTS 40962
#define NPOOL  10242
#define KNB    7
#define C1     64
#define C2     64
#define C3     128
#define INCH   8
#define OUTCH  64
#define NBATCH 8

#define KW2    (C1 * KNB)        // 448 : GEMM K for conv2 (neighbor-major)
#define KFC    (C3 * NPOOL)      // 1310976 : GEMM K for fc
#define KS_FC  (KFC / 4)         // 327744 k-steps of 4
#define NWAVE_FC 1024
#define PS_FC  ((KS_FC + NWAVE_FC - 1) / NWAVE_FC)   // 321 steps/wave

// W2[o][c][k] -> W2perm[o][k*64 + c]  (neighbor-major K so B-frags are
// contiguous float2 reads from each gathered h1 row)
__global__ __launch_bounds__(256) void perm_w2_kernel(const float* __restrict__ W2,
                                                      float* __restrict__ W2p) {
  int t = blockIdx.x * 256 + threadIdx.x;
  if (t >= C2 * KW2) return;
  int o = t / KW2, r = t % KW2;
  int k = r / C1, c = r % C1;
  W2p[t] = W2[(o * C1 + c) * KNB + k];
}

// conv1: h1[b][v][o] = relu(b1[o] + sum_c W1[o][c] * x[b][c][v])
__global__ __launch_bounds__(256) void conv1_kernel(const float* __restrict__ x,
                                                    const float* __restrict__ W1,
                                                    const float* __restrict__ b1,
                                                    float* __restrict__ h1) {
  __shared__ float W1s[C1 * INCH];
  __shared__ float b1s[C1];
  int tid = threadIdx.x;
  for (int i = tid; i < C1 * INCH; i += 256) W1s[i] = W1[i];
  if (tid < C1) b1s[tid] = b1[tid];
  __syncthreads();

  int v = blockIdx.x * 256 + tid;
  int b = blockIdx.y;
  if (v >= NVERTS) return;

  float xv[INCH];
#pragma unroll
  for (int c = 0; c < INCH; ++c)
    xv[c] = x[((size_t)b * INCH + c) * NVERTS + v];

  float* dst = h1 + ((size_t)b * NVERTS + v) * C1;
#pragma unroll
  for (int o4 = 0; o4 < C1; o4 += 4) {
    float4 r;
    float* rp = &r.x;
#pragma unroll
    for (int j = 0; j < 4; ++j) {
      int o = o4 + j;
      float a = b1s[o];
#pragma unroll
      for (int c = 0; c < INCH; ++c) a += W1s[o * INCH + c] * xv[c];
      rp[j] = fmaxf(a, 0.0f);
    }
    *reinterpret_cast<float4*>(dst + o4) = r;
  }
}

// conv2: per wave, one 16-vertex tile. D[64 x 16] = W2perm[64 x 448] * Bgathered[448 x 16]
__global__ __launch_bounds__(256) void conv2_wmma_kernel(const float* __restrict__ h1,
                                                         const int* __restrict__ nebs,
                                                         const float* __restrict__ W2p,
                                                         const float* __restrict__ b2,
                                                         float* __restrict__ h2) {
  const int NT = (NVERTS + 15) / 16;   // 2561
  int tile = blockIdx.x * 8 + (threadIdx.x >> 5);
  if (tile >= NT) return;              // wave-uniform exit, EXEC stays all-ones
  int b = blockIdx.y;
  int lane = threadIdx.x & 31;
  int lm = lane & 15, kg = lane >> 4;
  int v = tile * 16 + lm;
  int vc = v < NVERTS ? v : NVERTS - 1;

  const float* hrow[KNB];
#pragma unroll
  for (int k = 0; k < KNB; ++k) {
    int nb = nebs[vc * KNB + k];
    hrow[k] = h1 + ((size_t)b * NVERTS + (size_t)nb) * C1;
  }
  const float* arow[4];
#pragma unroll
  for (int mt = 0; mt < 4; ++mt)
    arow[mt] = W2p + (size_t)(mt * 16 + lm) * KW2;

  v8f acc[4] = {};
#pragma unroll
  for (int k = 0; k < KNB; ++k) {
    const float* bp = hrow[k];
#pragma unroll
    for (int s = 0; s < 16; ++s) {
      int c0 = k * C1 + s * 4 + kg * 2;      // K pair for this lane (A and B share it)
      v2f bf = *reinterpret_cast<const v2f*>(bp + (s * 4 + kg * 2));
#pragma unroll
      for (int mt = 0; mt < 4; ++mt) {
        v2f af = *reinterpret_cast<const v2f*>(arow[mt] + c0);
        acc[mt] = __builtin_amdgcn_wmma_f32_16x16x4_f32(
            false, af, false, bf, (short)0, acc[mt], false, false);
      }
    }
  }

  if (v < NVERTS) {
    float* dst = h2 + ((size_t)b * NVERTS + v) * C2;
#pragma unroll
    for (int mt = 0; mt < 4; ++mt) {
      int obase = mt * 16 + kg * 8;          // C/D layout: VGPR i -> M = 8*kg + i
      float4 r0, r1;
      float* p0 = &r0.x;
      float* p1 = &r1.x;
#pragma unroll
      for (int i = 0; i < 4; ++i) {
        p0[i] = fmaxf(acc[mt][i]     + b2[obase + i],     0.0f);
        p1[i] = fmaxf(acc[mt][i + 4] + b2[obase + i + 4], 0.0f);
      }
      *reinterpret_cast<float4*>(dst + obase)     = r0;
      *reinterpret_cast<float4*>(dst + obase + 4) = r1;
    }
  }
}

// pool: p[b][i][c] = max_k h2[b][nebs[i*7+k]][c]
__global__ __launch_bounds__(256) void pool_kernel(const float* __restrict__ h2,
                                                   const int* __restrict__ nebs,
                                                   float* __restrict__ p) {
  int t = blockIdx.x * 256 + threadIdx.x;
  if (t >= NBATCH * NPOOL * 16) return;
  int sub = t & 15;
  unsigned q = (unsigned)t >> 4;
  int i = q % NPOOL;
  int b = q / NPOOL;

  float4 m = make_float4(-1e30f, -1e30f, -1e30f, -1e30f);
#pragma unroll
  for (int k = 0; k < KNB; ++k) {
    int nb = nebs[i * KNB + k];
    const float4 g = *reinterpret_cast<const float4*>(
        h2 + ((size_t)b * NVERTS + (size_t)nb) * C2 + sub * 4);
    m.x = fmaxf(m.x, g.x); m.y = fmaxf(m.y, g.y);
    m.z = fmaxf(m.z, g.z); m.w = fmaxf(m.w, g.w);
  }
  *reinterpret_cast<float4*>(p + ((size_t)b * NPOOL + i) * C2 + sub * 4) = m;
}

// conv3: D[128 x 16] = W3[128 x 64] * p-tile[64 x 16]; store channel-major h3c[b][o][i]
__global__ __launch_bounds__(256) void conv3_wmma_kernel(const float* __restrict__ p,
                                                         const float* __restrict__ W3,
                                                         const float* __restrict__ b3,
                                                         float* __restrict__ h3c) {
  const int NT = (NPOOL + 15) / 16;    // 641
  int tile = blockIdx.x * 8 + (threadIdx.x >> 5);
  if (tile >= NT) return;
  int b = blockIdx.y;
  int lane = threadIdx.x & 31;
  int lm = lane & 15, kg = lane >> 4;
  int v = tile * 16 + lm;
  int vc = v < NPOOL ? v : NPOOL - 1;

  const float* brow = p + ((size_t)b * NPOOL + vc) * C2;
  v8f acc[8] = {};
#pragma unroll
  for (int s = 0; s < 16; ++s) {
    int c0 = s * 4 + kg * 2;
    v2f bf = *reinterpret_cast<const v2f*>(brow + c0);
#pragma unroll
    for (int mt = 0; mt < 8; ++mt) {
      v2f af = *reinterpret_cast<const v2f*>(W3 + (size_t)(mt * 16 + lm) * C2 + c0);
      acc[mt] = __builtin_amdgcn_wmma_f32_16x16x4_f32(
          false, af, false, bf, (short)0, acc[mt], false, false);
    }
  }

#pragma unroll
  for (int mt = 0; mt < 8; ++mt) {
#pragma unroll
    for (int i = 0; i < 8; ++i) {
      int o = mt * 16 + kg * 8 + i;
      float val = fmaxf(acc[mt][i] + b3[o], 0.0f);
      if (v < NPOOL)
        h3c[((size_t)b * C3 + o) * NPOOL + v] = val;   // lanes 0-15 consecutive v: coalesced
    }
  }
}

// fc: out[8 x 64] = flat[8 x 1.31M] * Wfc^T. Pad batch 8->16 cols (upper lanes scaled by 0
// so EXEC stays all-ones for WMMA). Each wave owns a contiguous K-chunk; partials reduced
// deterministically in fc_reduce (no float atomics -> bitwise stable across replays).
__global__ __launch_bounds__(256) void fc_wmma_kernel(const float* __restrict__ h3c,
                                                      const float* __restrict__ Wfc,
                                                      float* __restrict__ part) {
  int w = blockIdx.x * 8 + (threadIdx.x >> 5);
  int lane = threadIdx.x & 31;
  int lm = lane & 15, kg = lane >> 4;

  const float* bptr = h3c + (size_t)(lm & 7) * KFC;
  float bscale = (lm < 8) ? 1.0f : 0.0f;
  const float* aptr[4];
#pragma unroll
  for (int mt = 0; mt < 4; ++mt)
    aptr[mt] = Wfc + (size_t)(mt * 16 + lm) * KFC;

  v8f acc[4] = {};
  int s0 = w * PS_FC;
  int s1 = s0 + PS_FC; if (s1 > KS_FC) s1 = KS_FC;
  for (int s = s0; s < s1; ++s) {
    int r = s * 4 + kg * 2;
    v2f bf = *reinterpret_cast<const v2f*>(bptr + r);
    bf = bf * bscale;                      // zero the 8 padded columns
#pragma unroll
    for (int mt = 0; mt < 4; ++mt) {
      v2f af = *reinterpret_cast<const v2f*>(aptr[mt] + r);
      acc[mt] = __builtin_amdgcn_wmma_f32_16x16x4_f32(
          false, af, false, bf, (short)0, acc[mt], false, false);
    }
  }

#pragma unroll
  for (int mt = 0; mt < 4; ++mt) {
#pragma unroll
    for (int i = 0; i < 8; ++i) {
      int o = mt * 16 + kg * 8 + i;
      part[((size_t)w * OUTCH + o) * 16 + lm] = acc[mt][i];
    }
  }
}

__global__ __launch_bounds__(512) void fc_reduce_kernel(const float* __restrict__ part,
                                                        const float* __restrict__ bfc,
                                                        float* __restrict__ out) {
  int t = threadIdx.x;
  if (t >= NBATCH * OUTCH) return;
  int b = t >> 6, o = t & 63;
  float s = 0.0f;
  for (int w = 0; w < NWAVE_FC; ++w)
    s += part[((size_t)w * OUTCH + o) * 16 + b];
  out[b * OUTCH + o] = s + bfc[o];
}

extern "C" void kernel_launch(void* const* d_in, const int* in_sizes, int n_in,
                              void* d_out, int out_size, void* d_ws, size_t ws_size,
                              hipStream_t stream) {
  const float* x    = (const float*)d_in[0];
  const int*   nebs = (const int*)d_in[1];
  const float* W1   = (const float*)d_in[2];
  const float* b1   = (const float*)d_in[3];
  const float* W2   = (const float*)d_in[4];
  const float* b2   = (const float*)d_in[5];
  const float* W3   = (const float*)d_in[6];
  const float* b3   = (const float*)d_in[7];
  const float* Wfc  = (const float*)d_in[8];
  const float* bfc  = (const float*)d_in[9];
  float* out = (float*)d_out;
  float* ws  = (float*)d_ws;

  // Workspace layout (floats), with dead-buffer reuse:
  const size_t SZ_H1 = (size_t)NBATCH * NVERTS * C1;   // 20,972,544
  const size_t SZ_H2 = (size_t)NBATCH * NVERTS * C2;   // 20,972,544
  const size_t SZ_P  = (size_t)NBATCH * NPOOL * C2;    //  5,243,904
  float* h1   = ws;
  float* h2   = ws + SZ_H1;
  float* pbuf = ws + SZ_H1 + SZ_H2;
  float* w2p  = ws + SZ_H1 + SZ_H2 + SZ_P;             // 28,672 floats
  float* h3c  = h1;   // h1 dead after conv2 (needs 10,487,808 <= SZ_H1)
  float* part = h2;   // h2 dead after pool  (needs  1,048,576 <= SZ_H2)

  perm_w2_kernel<<<(C2 * KW2 + 255) / 256, 256, 0, stream>>>(W2, w2p);
  conv1_kernel<<<dim3((NVERTS + 255) / 256, NBATCH), 256, 0, stream>>>(x, W1, b1, h1);
  conv2_wmma_kernel<<<dim3(((NVERTS + 15) / 16 + 7) / 8, NBATCH), 256, 0, stream>>>(
      h1, nebs, w2p, b2, h2);
  pool_kernel<<<(NBATCH * NPOOL * 16 + 255) / 256, 256, 0, stream>>>(h2, nebs, pbuf);
  conv3_wmma_kernel<<<dim3(((NPOOL + 15) / 16 + 7) / 8, NBATCH), 256, 0, stream>>>(
      pbuf, W3, b3, h3c);
  fc_wmma_kernel<<<NWAVE_FC / 8, 256, 0, stream>>>(h3c, Wfc, part);
  fc_reduce_kernel<<<1, 512, 0, stream>>>(part, bfc, out);
}